// RotatedIOULoss_61151744360589
// MI455X (gfx1250) — compile-verified
//
#include <hip/hip_runtime.h>
#include <math.h>
#include <utility>

#define BLK 256
#define RIOU_EPS 1e-8f
#define RIOU_TOL 1e-6f

typedef __attribute__((ext_vector_type(2))) float v2f;
typedef __attribute__((ext_vector_type(8))) float v8f;

// ---------------------------------------------------------------------------
// static_for: indices are std::integral_constant -> every array subscript is a
// frontend compile-time constant -> mem2reg promotes all slot arrays to SSA
// registers (no promote-alloca vector + cndmask-waterfall blowup).
// ---------------------------------------------------------------------------
template <typename F, int... Is>
__device__ __forceinline__ void sf_impl(F&& f, std::integer_sequence<int, Is...>) {
    (f(std::integral_constant<int, Is>{}), ...);
}
template <int N, typename F>
__device__ __forceinline__ void static_for(F&& f) {
    sf_impl(f, std::make_integer_sequence<int, N>{});
}

// ---------------------------------------------------------------------------
// Compile-time Batcher odd-even merge sorting network for 24 slots (padded to
// 32; all comparators ascending, so comparators touching the virtual +inf pad
// slots 24..31 never act and are dropped at generation time).
// ---------------------------------------------------------------------------
constexpr int NSLOT = 24;
constexpr int NPAD  = 32;

constexpr int ce_count() {
    int c = 0;
    for (int p = 1; p < NPAD; p <<= 1)
        for (int k = p; k >= 1; k >>= 1)
            for (int j = k % p; j + k < NPAD; j += 2 * k)
                for (int i = 0; i < k; ++i) {
                    int a = i + j, b = i + j + k;
                    if (b < NPAD && (a / (2 * p) == b / (2 * p)) && b < NSLOT)
                        ++c;
                }
    return c;
}
constexpr int NCE = ce_count();

struct CETab { unsigned char a[NCE]; unsigned char b[NCE]; };

constexpr CETab make_ce() {
    CETab t{};
    int c = 0;
    for (int p = 1; p < NPAD; p <<= 1)
        for (int k = p; k >= 1; k >>= 1)
            for (int j = k % p; j + k < NPAD; j += 2 * k)
                for (int i = 0; i < k; ++i) {
                    int a = i + j, b = i + j + k;
                    if (b < NPAD && (a / (2 * p) == b / (2 * p)) && b < NSLOT) {
                        t.a[c] = (unsigned char)a;
                        t.b[c] = (unsigned char)b;
                        ++c;
                    }
                }
    return t;
}
constexpr CETab CES = make_ce();

// ---------------------------------------------------------------------------
// Helpers
// ---------------------------------------------------------------------------
__device__ __forceinline__ float rcp_fast(float x) {   // v_rcp_f32, 1 ulp
    return __builtin_amdgcn_rcpf(x);
}

struct BoxFrame { float ax, ay, abx, aby, adx, ady, iab, iad; };

__device__ __forceinline__ BoxFrame make_frame(const float (&cx)[4],
                                               const float (&cy)[4]) {
    BoxFrame f;
    f.ax = cx[0];           f.ay = cy[0];
    f.abx = cx[1] - cx[0];  f.aby = cy[1] - cy[0];
    f.adx = cx[3] - cx[0];  f.ady = cy[3] - cy[0];
    f.iab = rcp_fast(f.abx * f.abx + f.aby * f.aby);
    f.iad = rcp_fast(f.adx * f.adx + f.ady * f.ady);
    return f;
}

__device__ __forceinline__ bool in_frame(float px, float py, const BoxFrame& f) {
    float amx = px - f.ax, amy = py - f.ay;
    float pab = (f.abx * amx + f.aby * amy) * f.iab;
    float pad = (f.adx * amx + f.ady * amy) * f.iad;
    return (pab > -RIOU_TOL) && (pab < 1.0f + RIOU_TOL) &&
           (pad > -RIOU_TOL) && (pad < 1.0f + RIOU_TOL);
}

// Pseudo-angle: strictly monotone with atan2(y,x) on (-pi, pi], range [-2, 2].
// Identical sort order => identical polygon => identical shoelace area.
__device__ __forceinline__ float pseudo_angle(float x, float y) {
    float r = rcp_fast(fabsf(x) + fabsf(y) + 1e-30f);
    return copysignf(1.0f - x * r, y);
}

__device__ __forceinline__ void box_corners(const float (&b)[7],
                                            float (&cx)[4], float (&cy)[4]) {
    float x = b[0], y = b[1], w = b[3], h = b[4], a = b[6];
    float s, c;
    __sincosf(a, &s, &c);
    static_for<4>([&](auto I) {
        constexpr int k = decltype(I)::value;
        constexpr float hx = (k == 0 || k == 3) ? 0.5f : -0.5f;
        constexpr float hy = (k == 0 || k == 1) ? 0.5f : -0.5f;
        float xx = hx * w, yy = hy * h;
        cx[k] = xx * c - yy * s + x;
        cy[k] = xx * s + yy * c + y;
    });
}

// ---------------------------------------------------------------------------
// Stage 1: per-pair loss, per-block partial sum. Inputs staged with CDNA5
// async global->LDS DMA (ASYNCcnt path); all geometry branch-free in VGPRs.
// ---------------------------------------------------------------------------
__global__ void __launch_bounds__(BLK)
riou_main(const float* __restrict__ pred, const float* __restrict__ tgt,
          float* __restrict__ partial, int N) {
    __shared__ float smem[2 * 7 * BLK];   // 14,336 B staging only
    __shared__ float wsum[BLK / 32];

    const int tid = threadIdx.x;
    const long long base = (long long)blockIdx.x * BLK;
    const long long lastDword = (long long)N * 7 - 1;

    unsigned long long pbase = (unsigned long long)(const void*)pred;
    unsigned long long tbase = (unsigned long long)(const void*)tgt;

    // --- async coalesced staging: 7 dword-waves each for pred and target ---
    // NOTE: plain unrolled loop (not static_for): clang forbids inline-asm
    // operands that name by-ref-captured enclosing-function locals in lambdas.
#pragma unroll
    for (int r = 0; r < 7; ++r) {
        long long idx = base * 7 + (long long)r * BLK + tid;
        if (idx > lastDword) idx = lastDword;            // clamp, no divergence
        unsigned goff = (unsigned)(idx * 4);             // < 8 MB, fits 32-bit
        unsigned lp = (unsigned)(size_t)&smem[r * BLK + tid];
        unsigned lt = (unsigned)(size_t)&smem[7 * BLK + r * BLK + tid];
        asm volatile("global_load_async_to_lds_b32 %0, %1, %2 offset:0"
                     :: "v"(lp), "v"(goff), "s"(pbase) : "memory");
        asm volatile("global_load_async_to_lds_b32 %0, %1, %2 offset:0"
                     :: "v"(lt), "v"(goff), "s"(tbase) : "memory");
    }
    asm volatile("s_wait_asynccnt 0" ::: "memory");
    __syncthreads();

    float p[7], q[7];
    static_for<7>([&](auto J) {
        constexpr int j = decltype(J)::value;
        p[j] = smem[tid * 7 + j];               // stride 7: bank-conflict-free
        q[j] = smem[7 * BLK + tid * 7 + j];
    });

    // --- corners ---
    float c1x[4], c1y[4], c2x[4], c2y[4];
    box_corners(p, c1x, c1y);
    box_corners(q, c2x, c2y);
    BoxFrame f1 = make_frame(c1x, c1y);
    BoxFrame f2 = make_frame(c2x, c2y);

    // --- 24 candidate vertices: 4 + 4 corners, 16 edge-edge intersections ---
    float kx[NSLOT], ky[NSLOT], key[NSLOT];
    int nv = 0;
    float sx = 0.0f, sy = 0.0f;
    bool mm[NSLOT];

    static_for<4>([&](auto K) {
        constexpr int k = decltype(K)::value;
        kx[k] = c1x[k]; ky[k] = c1y[k];
        mm[k] = in_frame(c1x[k], c1y[k], f2);
        kx[4 + k] = c2x[k]; ky[4 + k] = c2y[k];
        mm[4 + k] = in_frame(c2x[k], c2y[k], f1);
    });
    static_for<16>([&](auto S) {
        constexpr int s = decltype(S)::value;
        constexpr int k = s >> 2, j = s & 3;
        float x1 = c1x[k], y1 = c1y[k];
        float x2 = c1x[(k + 1) & 3], y2 = c1y[(k + 1) & 3];
        float x3 = c2x[j], y3 = c2y[j];
        float x4 = c2x[(j + 1) & 3], y4 = c2y[(j + 1) & 3];
        float num  = (x1 - x2) * (y3 - y4) - (y1 - y2) * (x3 - x4);
        float dent = (x1 - x3) * (y3 - y4) - (y1 - y3) * (x3 - x4);
        float denu = (x1 - x2) * (y1 - y3) - (y1 - y2) * (x1 - x3);
        bool  nz   = (num != 0.0f);
        float rs   = rcp_fast(nz ? num : 1.0f);
        float tm   = nz ? dent * rs : -1.0f;
        float um   = nz ? -denu * rs : -1.0f;
        bool  m = (tm > 0.0f) && (tm < 1.0f) && (um > 0.0f) && (um < 1.0f);
        float tt = dent * rcp_fast(num + RIOU_EPS);
        kx[8 + s] = x1 + tt * (x2 - x1);
        ky[8 + s] = y1 + tt * (y2 - y1);
        mm[8 + s] = m;
    });

    // --- mean over valid ---
    static_for<NSLOT>([&](auto I) {
        constexpr int i = decltype(I)::value;
        float f = mm[i] ? 1.0f : 0.0f;
        nv += mm[i] ? 1 : 0;
        sx += kx[i] * f;
        sy += ky[i] * f;
    });
    float inv = 1.0f / fmaxf((float)nv, 1.0f);   // precise, 1 per thread
    float mx = sx * inv, my = sy * inv;

    // --- centered/masked coords + pseudo-angle keys ---
    static_for<NSLOT>([&](auto I) {
        constexpr int i = decltype(I)::value;
        float f = mm[i] ? 1.0f : 0.0f;
        float vx = (kx[i] - mx) * f;
        float vy = (ky[i] - my) * f;
        kx[i] = vx; ky[i] = vy;
        key[i] = mm[i] ? pseudo_angle(vx, vy) : 1.0e9f;
    });

    // --- branch-free sorting network on (key, x, y) triples, pure VGPR SSA ---
    static_for<NCE>([&](auto T) {
        constexpr int a = CES.a[decltype(T)::value];
        constexpr int b = CES.b[decltype(T)::value];
        bool sw = key[a] > key[b];
        float t0 = sw ? key[b] : key[a];
        float t1 = sw ? key[a] : key[b];
        key[a] = t0; key[b] = t1;
        float t2 = sw ? kx[b] : kx[a];
        float t3 = sw ? kx[a] : kx[b];
        kx[a] = t2; kx[b] = t3;
        float t4 = sw ? ky[b] : ky[a];
        float t5 = sw ? ky[a] : ky[b];
        ky[a] = t4; ky[b] = t5;
    });

    // --- shoelace: open chain (invalid slots are (0,0) -> zero terms),
    //     plus closing edge (vertex nv-1 -> vertex 0) via select-scan ---
    float cr = 0.0f;
    static_for<NSLOT - 1>([&](auto I) {
        constexpr int i = decltype(I)::value;
        cr += kx[i] * ky[i + 1] - ky[i] * kx[i + 1];
    });
    float lx = 0.0f, ly = 0.0f;
    static_for<NSLOT>([&](auto I) {
        constexpr int i = decltype(I)::value;
        bool sel = (i == nv - 1);
        lx = sel ? kx[i] : lx;
        ly = sel ? ky[i] : ly;
    });
    cr += lx * ky[0] - ly * kx[0];
    float area = fabsf(cr) * 0.5f;

    // --- z overlap + IoU loss ---
    float zmax1 = p[2] + 0.5f * p[5], zmin1 = p[2] - 0.5f * p[5];
    float zmax2 = q[2] + 0.5f * q[5], zmin2 = q[2] - 0.5f * q[5];
    float zo = fmaxf(fminf(zmax1, zmax2) - fmaxf(zmin1, zmin2), 0.0f);
    float inter3d = area * zo;
    float v1 = p[3] * p[4] * p[5];
    float v2 = q[3] * q[4] * q[5];
    float u3d = v1 + v2 - inter3d;
    float loss = -__logf((inter3d + 1.0f) / (u3d + 1.0f));  // precise ratio
    if (base + tid >= N) loss = 0.0f;

    // --- deterministic wave32 + block reduction ---
#pragma unroll
    for (int o = 16; o > 0; o >>= 1) loss += __shfl_down(loss, o, 32);
    if ((tid & 31) == 0) wsum[tid >> 5] = loss;
    __syncthreads();
    if (tid == 0) {
        float s = 0.0f;
        static_for<BLK / 32>([&](auto I) { s += wsum[decltype(I)::value]; });
        partial[blockIdx.x] = s;
    }
}

// ---------------------------------------------------------------------------
// Stage 2: deterministic final reduction of block partials using a real
// V_WMMA_F32_16X16X4_F32. With B = ones(4x16), D[m][n] = sum_k A[m][k]:
// lane layout gives r[m] = partial[m] + partial[m+16] replicated over n;
// lanes 0-15 then hold sum(r[0..7]) and lanes 16-31 hold sum(r[8..15]).
// One wave, EXEC all ones (WMMA requirement), exact f32 arithmetic.
// ---------------------------------------------------------------------------
__global__ void __launch_bounds__(32)
riou_reduce(const float* __restrict__ part, float* __restrict__ out, int M) {
    int lane = threadIdx.x;
    float s = 0.0f;
    for (int i = lane; i < M; i += 32) s += part[i];

    v2f a; a.x = s;    a.y = 0.0f;   // A: 16x4, K=1/3 columns zero
    v2f b; b.x = 1.0f; b.y = 1.0f;   // B: 4x16 all ones (layout-independent)
    v8f c = {};
    v8f d = __builtin_amdgcn_wmma_f32_16x16x4_f32(
        /*neg_a=*/false, a, /*neg_b=*/false, b,
        /*c_mod=*/(short)0, c, /*reuse_a=*/false, /*reuse_b=*/false);

    float t = d[0] + d[1] + d[2] + d[3] + d[4] + d[5] + d[6] + d[7];
    float total = t + __shfl(t, lane ^ 16, 32);
    if (lane == 0) out[0] = total;
}

// ---------------------------------------------------------------------------
extern "C" void kernel_launch(void* const* d_in, const int* in_sizes, int n_in,
                              void* d_out, int out_size, void* d_ws, size_t ws_size,
                              hipStream_t stream) {
    const float* pred = (const float*)d_in[0];
    const float* tgt  = (const float*)d_in[1];
    int N = in_sizes[0] / 7;
    int nblk = (N + BLK - 1) / BLK;
    float* partial = (float*)d_ws;   // nblk floats of scratch

    riou_main<<<nblk, BLK, 0, stream>>>(pred, tgt, partial, N);
    riou_reduce<<<1, 32, 0, stream>>>(partial, (float*)d_out, nblk);
}